// TreeDecoder_36687610642488
// MI455X (gfx1250) — compile-verified
//
#include <hip/hip_runtime.h>
#include <hip/hip_bf16.h>
#include <math.h>

typedef __attribute__((ext_vector_type(2))) float v2f;
typedef __attribute__((ext_vector_type(8))) float v8f;

#define WMMA_F32(a, b, c) \
    __builtin_amdgcn_wmma_f32_16x16x4_f32(false, (a), false, (b), (short)0, (c), false, false)

__device__ __forceinline__ float mishf(float x) {
    // mish(x) = x * tanh(softplus(x)); softplus computed overflow-safely
    float sp = (x > 20.0f) ? x : log1pf(expf(x));
    return x * tanhf(sp);
}

// ---------------------------------------------------------------------------
// Kernel 1: x0 = mish(BN(trees @ W_lin^T + b_lin)), output (B, 4096) f32.
// Grid (4096/64, 256/16); block 128 = 4 waves; one 16x16 tile per wave.
// ---------------------------------------------------------------------------
__global__ __launch_bounds__(128) void k_linear(
    const float* __restrict__ trees, const float* __restrict__ Wl,
    const float* __restrict__ bl, const float* __restrict__ gam,
    const float* __restrict__ bet, const float* __restrict__ mu,
    const float* __restrict__ var, float* __restrict__ out)
{
    const int lane = threadIdx.x & 31;
    const int wave = threadIdx.x >> 5;
    const int n0 = blockIdx.x * 64 + wave * 16;   // LIN tile
    const int m0 = blockIdx.y * 16;               // batch tile
    const int l15 = lane & 15;
    const int hi  = lane >> 4;                    // half-wave selects K 0/2
    const int kb  = hi * 2;

    const float* arow = trees + (size_t)(m0 + l15) * 512 + kb; // A: lane = M
    const float* brow = Wl + (size_t)(n0 + l15) * 512 + kb;    // B[k][n]=Wl[n][k]

    v8f acc = {};
#pragma unroll 4
    for (int k0 = 0; k0 < 512; k0 += 4) {
        v2f a = *(const v2f*)(arow + k0);
        v2f b = *(const v2f*)(brow + k0);
        acc = WMMA_F32(a, b, acc);
    }

    const int nn = n0 + l15;
    const float bias = bl[nn];
    const float sc = rsqrtf(var[nn] + 1e-5f) * gam[nn];
    const float sh = bet[nn] - mu[nn] * sc;
#pragma unroll
    for (int r = 0; r < 8; ++r) {
        const int m = m0 + r + hi * 8;            // D: M = r (+8 for hi half)
        float t = acc[r] + bias;
        t = t * sc + sh;
        out[(size_t)m * 4096 + nn] = mishf(t);
    }
}

// ---------------------------------------------------------------------------
// Kernel 2: binary tree conv (pre-norm), templated on shapes.
// One WG = (batch, NT*16 columns). Gathered B tile staged once in LDS with
// K-paired layout so a lane's (K,K+1) B fragment is a single ds_load_b64;
// each A fragment (global b64) feeds NT WMMAs -> NT x less weight traffic.
//   out[b][o][j] = bc[o] + sum_{c,k} w[o][c][k]*x[b][c][idx[b][3(j-1)+k]],
//   j in 1..255; out[b][o][0] = 0.
// Grid (256/(16*NT), B); block 128 = 4 waves; m-tiles strided over waves.
// ---------------------------------------------------------------------------
template <int CIN, int COUT, int NT>
__global__ __launch_bounds__(128) void k_conv(
    const float* __restrict__ x, const int* __restrict__ idx,
    const float* __restrict__ w, const float* __restrict__ bc,
    float* __restrict__ y)
{
    constexpr int Kt    = 3 * CIN;
    constexpr int NCOLS = 16 * NT;
    __shared__ int sidx[3 * NCOLS];
    extern __shared__ float Bl[];                 // [Kt/2][NCOLS][2] K-paired

    const int b  = blockIdx.y;
    const int j0 = blockIdx.x * NCOLS;
    const float* xb = x + (size_t)b * CIN * 256;
    const int*   ib = idx + (size_t)b * 765;

    // preload the 3*NCOLS contiguous indices this WG needs
    {
        const int base = 3 * (j0 - 1);            // relative idx = 3n + kk
        for (int t = threadIdx.x; t < 3 * NCOLS; t += blockDim.x) {
            const int g = base + t;
            sidx[t] = (g >= 0) ? ib[g] : 0;       // g<0 only feeds the zero col
        }
    }
    __syncthreads();

    // cooperative gather of the B tile into LDS (K-paired layout)
    for (int t = threadIdx.x; t < Kt * NCOLS; t += blockDim.x) {
        const int K = t / NCOLS;                  // row = c*3 + kk
        const int n = t - K * NCOLS;
        float v = 0.0f;
        if (j0 + n != 0) {
            const int c  = K / 3;
            const int kk = K - 3 * c;
            v = xb[c * 256 + sidx[3 * n + kk]];
        }
        Bl[(K >> 1) * (NCOLS * 2) + n * 2 + (K & 1)] = v;
    }
    __syncthreads();

    const int lane = threadIdx.x & 31;
    const int wave = threadIdx.x >> 5;
    const int l15 = lane & 15;
    const int hi  = lane >> 4;
    const int kb  = hi * 2;
    constexpr int nMT = COUT >> 4;

    for (int mt = wave; mt < nMT; mt += 4) {      // uniform per wave: EXEC all-1s
        const int o0 = mt * 16;
        const float* arow = w + (size_t)(o0 + l15) * Kt + kb; // w: (COUT, Kt)
        v8f acc[NT];
#pragma unroll
        for (int nt = 0; nt < NT; ++nt) acc[nt] = (v8f){};

#pragma unroll 4
        for (int k0 = 0; k0 < Kt; k0 += 4) {
            const v2f a = *(const v2f*)(arow + k0);
            const int prow = (k0 >> 1) + hi;      // pair row of (k0+kb, +1)
#pragma unroll
            for (int nt = 0; nt < NT; ++nt) {
                const v2f bf = *(const v2f*)&Bl[prow * (NCOLS * 2) +
                                                (nt * 16 + l15) * 2];
                acc[nt] = WMMA_F32(a, bf, acc[nt]);
            }
        }

#pragma unroll
        for (int nt = 0; nt < NT; ++nt) {
            const int j = j0 + nt * 16 + l15;
#pragma unroll
            for (int r = 0; r < 8; ++r) {
                const int m = o0 + r + hi * 8;
                const float v = (j == 0) ? 0.0f : (acc[nt][r] + bc[m]);
                y[((size_t)b * COUT + m) * 256 + j] = v;
            }
        }
    }
}

// ---------------------------------------------------------------------------
// Kernel 3: per-batch tree_layer_norm (ddof=1, /(std+1e-5)) + optional mish,
// in place. One block per batch, two passes over C*256 elements.
// ---------------------------------------------------------------------------
__global__ __launch_bounds__(256) void k_norm(float* __restrict__ y, int C, int act)
{
    __shared__ float s0[256], s1[256];
    const int b = blockIdx.x;
    const int N = C * 256;
    float* yb = y + (size_t)b * N;

    float sum = 0.0f, sq = 0.0f;
    for (int i = threadIdx.x; i < N; i += 256) {
        const float v = yb[i];
        sum += v; sq += v * v;
    }
    s0[threadIdx.x] = sum; s1[threadIdx.x] = sq;
    __syncthreads();
    for (int s = 128; s > 0; s >>= 1) {
        if (threadIdx.x < s) {
            s0[threadIdx.x] += s0[threadIdx.x + s];
            s1[threadIdx.x] += s1[threadIdx.x + s];
        }
        __syncthreads();
    }
    const float mean = s0[0] / (float)N;
    const float var  = (s1[0] - (float)N * mean * mean) / (float)(N - 1);
    const float sd   = sqrtf(fmaxf(var, 0.0f));
    const float scl  = 1.0f / (sd + 1e-5f);

    for (int i = threadIdx.x; i < N; i += 256) {
        float v = (yb[i] - mean) * scl;
        if (act) v = mishf(v);
        yb[i] = v;
    }
}

// ---------------------------------------------------------------------------
// Host side
// ---------------------------------------------------------------------------
template <int CIN, int COUT, int NT>
static inline void launch_conv(const float* src, const int* idx, const float* w,
                               const float* bc, float* dst, hipStream_t stream)
{
    const size_t lds = (size_t)(3 * CIN) * 16 * NT * sizeof(float);
    k_conv<CIN, COUT, NT><<<dim3(256 / (16 * NT), 256), 128, lds, stream>>>(
        src, idx, w, bc, dst);
}

extern "C" void kernel_launch(void* const* d_in, const int* in_sizes, int n_in,
                              void* d_out, int out_size, void* d_ws, size_t ws_size,
                              hipStream_t stream)
{
    const float* trees   = (const float*)d_in[0];
    const int*   indexes = (const int*)  d_in[1];   // (256, 765, 1) int32
    const float* W_lin   = (const float*)d_in[2];
    const float* b_lin   = (const float*)d_in[3];
    const float* bn_g    = (const float*)d_in[4];
    const float* bn_b    = (const float*)d_in[5];
    const float* bn_m    = (const float*)d_in[6];
    const float* bn_v    = (const float*)d_in[7];
    const float* W[6]; const float* BC[6];
    for (int i = 0; i < 6; ++i) {
        W[i]  = (const float*)d_in[8 + 2 * i];
        BC[i] = (const float*)d_in[9 + 2 * i];
    }

    // ws layout: region0 (64 MB): x0 / a1 / a3 ; region1 (128 MB): a0 / a2 / a4
    float* ws0 = (float*)d_ws;
    float* ws1 = ws0 + ((size_t)64 << 20) / sizeof(float);
    float* out = (float*)d_out;

    // Stage 0: x0 = mish(BN(trees @ W_lin^T + b_lin)) -> (256, 16, 256) in ws0
    k_linear<<<dim3(64, 16), 128, 0, stream>>>(trees, W_lin, b_lin, bn_g, bn_b,
                                               bn_m, bn_v, ws0);

    // Layer 0: 16 -> 32 (no mish)
    launch_conv<16, 32, 4>(ws0, indexes, W[0], BC[0], ws1, stream);
    k_norm<<<dim3(256), 256, 0, stream>>>(ws1, 32, 0);
    // Layer 1: 32 -> 64
    launch_conv<32, 64, 4>(ws1, indexes, W[1], BC[1], ws0, stream);
    k_norm<<<dim3(256), 256, 0, stream>>>(ws0, 64, 1);
    // Layer 2: 64 -> 128
    launch_conv<64, 128, 4>(ws0, indexes, W[2], BC[2], ws1, stream);
    k_norm<<<dim3(256), 256, 0, stream>>>(ws1, 128, 1);
    // Layer 3: 128 -> 256   (LDS 96 KB)
    launch_conv<128, 256, 4>(ws1, indexes, W[3], BC[3], ws0, stream);
    k_norm<<<dim3(256), 256, 0, stream>>>(ws0, 256, 1);
    // Layer 4: 256 -> 512   (LDS 96 KB, NT=2)
    launch_conv<256, 512, 2>(ws0, indexes, W[4], BC[4], ws1, stream);
    k_norm<<<dim3(256), 256, 0, stream>>>(ws1, 512, 1);
    // Layer 5: 512 -> 512   (LDS 192 KB, NT=2), straight into d_out
    launch_conv<512, 512, 2>(ws1, indexes, W[5], BC[5], out, stream);
    k_norm<<<dim3(256), 256, 0, stream>>>(out, 512, 1);
}